// SRELoss_neg_38319698215598
// MI455X (gfx1250) — compile-verified
//
#include <hip/hip_runtime.h>
#include <hip/hip_bf16.h>

#define B_    32
#define C_    768
#define H_    32
#define W_    32
#define HW_   (H_ * W_)
#define NCLS_ 21
#define CROP_ 8
#define HC_   26
#define NF_   (HC_ * HC_)        // 676 candidate centers per batch
#define HALF_ ((unsigned)(B_ * NF_ / 2))  // 10816, threefry counter half
#define EPS_  1e-8f

typedef __attribute__((ext_vector_type(2))) float v2f;
typedef __attribute__((ext_vector_type(8))) float v8f;

// ---------------------------------------------------------------------------
// JAX threefry2x32 (key = (0,42)), matching jax.random.uniform(key(42), (B,676))
// ---------------------------------------------------------------------------
__device__ __forceinline__ unsigned rotl32(unsigned x, unsigned r) {
    return (x << r) | (x >> (32u - r));
}

__device__ __forceinline__ float jax_threefry_uniform(unsigned e) {
    unsigned j  = (e < HALF_) ? e : (e - HALF_);
    unsigned x0 = j;
    unsigned x1 = j + HALF_;
    unsigned ks[3] = {0u, 42u, 0u ^ 42u ^ 0x1BD11BDAu};
    const unsigned rotA[4] = {13u, 15u, 26u, 6u};
    const unsigned rotB[4] = {17u, 29u, 16u, 24u};
    x0 += ks[0]; x1 += ks[1];
#pragma unroll
    for (int i = 0; i < 5; ++i) {
        const unsigned* r = (i & 1) ? rotB : rotA;
#pragma unroll
        for (int q = 0; q < 4; ++q) { x0 += x1; x1 = rotl32(x1, r[q]); x1 ^= x0; }
        x0 += ks[(i + 1) % 3];
        x1 += ks[(i + 2) % 3] + (unsigned)(i + 1);
    }
    unsigned bits = (e < HALF_) ? x0 : x1;
    unsigned fb   = (bits >> 9) | 0x3f800000u;
    return __uint_as_float(fb) - 1.0f;
}

// ---------------------------------------------------------------------------
// Kernel 1: roi -> window sums -> valid -> threefry scores -> top-8 selection
// One block per batch image. Also zeroes the scalar output.
// ---------------------------------------------------------------------------
__global__ __launch_bounds__(256) void sre_select_kernel(
    const int* __restrict__ masks, int* __restrict__ sel, float* __restrict__ out)
{
    const int b = blockIdx.x;
    const int t = threadIdx.x;
    if (b == 0 && t == 0) out[0] = 0.0f;   // kernel-boundary ordering makes this safe

    __shared__ float roi[HW_];
    __shared__ float score[NF_];
    __shared__ float rv[256];
    __shared__ int   ri[256];

    const int* mB = masks + b * HW_;
    for (int i = t; i < HW_; i += 256) {
        int m = mB[i];
        roi[i] = (m == 255) ? 1.0f : (m >= 1 ? 2.0f : 0.0f);
    }
    __syncthreads();

    for (int f = t; f < NF_; f += 256) {
        int y = f / HC_, x = f % HC_;
        float ws = 0.0f;
#pragma unroll
        for (int dy = 0; dy < 6; ++dy)
#pragma unroll
            for (int dx = 0; dx < 6; ++dx)
                ws += roi[(y + dy) * W_ + (x + dx)];
        bool valid = (roi[(y + 3) * W_ + (x + 3)] == 1.0f) && (ws >= 43.2f);
        float u = jax_threefry_uniform((unsigned)(b * NF_ + f));
        score[f] = valid ? u : -1.0e9f;
    }
    __syncthreads();

    // 8 rounds of argmax; tie-break on lower index (matches lax.top_k ordering)
    for (int k = 0; k < CROP_; ++k) {
        float best = -3.0e9f;
        int   bi   = 0x7fffffff;
        for (int f = t; f < NF_; f += 256) {
            float v = score[f];
            if (v > best || (v == best && f < bi)) { best = v; bi = f; }
        }
        rv[t] = best; ri[t] = bi;
        __syncthreads();
        for (int s = 128; s > 0; s >>= 1) {
            if (t < s) {
                float v2 = rv[t + s]; int i2 = ri[t + s];
                if (v2 > rv[t] || (v2 == rv[t] && i2 < ri[t])) { rv[t] = v2; ri[t] = i2; }
            }
            __syncthreads();
        }
        if (t == 0) {
            int f = ri[0];
            int y = f / HC_, x = f % HC_;
            sel[b * CROP_ + k] = y * W_ + x;   // pixel offset into full 32x32 image
            score[f] = -3.0e9f;                // exclude from later rounds
        }
        __syncthreads();
    }
}

// ---------------------------------------------------------------------------
// Kernel 2: gather 8 columns, L2-normalize tokens + class tokens, fp32 WMMA
// GEMM (16x21 padded to 2x 16x16 tiles, K=768), weighted loss, atomic scalar.
// One wave (32 lanes) per batch. All fragment loads are branchless: padded
// rows/cols read a valid dummy address and are zeroed by their 0 inv-norm.
// ---------------------------------------------------------------------------
__global__ __launch_bounds__(32) void sre_gemm_kernel(
    const float* __restrict__ cls,     // (B, 21, 768)
    const float* __restrict__ pt,      // (B, 768, 32, 32)
    const float* __restrict__ labels,  // (B, 21)
    const int*   __restrict__ sel,     // (B, 8) pixel offsets
    float* __restrict__ out)
{
    const int b    = blockIdx.x;
    const int lane = threadIdx.x;

    __shared__ float invA[16];
    __shared__ float invB[32];
    __shared__ int   pOff[16];         // entries 8..15 stay 0 (valid dummy addr)
    __shared__ float labSh[32];

    const float* ptB  = pt  + (size_t)b * C_ * HW_;
    const float* clsB = cls + (size_t)b * NCLS_ * C_;

    if (lane < 16) { invA[lane] = 0.0f; pOff[lane] = 0; }
    invB[lane]  = 0.0f;
    labSh[lane] = (lane < NCLS_) ? labels[b * NCLS_ + lane] : 0.0f;
    __syncthreads();
    if (lane < CROP_) pOff[lane] = sel[b * CROP_ + lane];
    __syncthreads();

    // per-token inverse L2 norms (wave-cooperative reductions)
    for (int m = 0; m < CROP_; ++m) {
        int p   = pOff[m];
        float s = 0.0f;
        for (int c = lane; c < C_; c += 32) {
            float v = ptB[(size_t)c * HW_ + p];
            s += v * v;
        }
#pragma unroll
        for (int o = 16; o > 0; o >>= 1) s += __shfl_xor(s, o, 32);
        if (lane == 0) invA[m] = 1.0f / fmaxf(sqrtf(s), EPS_);
    }
    for (int n = 0; n < NCLS_; ++n) {
        float s = 0.0f;
        for (int c = lane; c < C_; c += 32) {
            float v = clsB[n * C_ + c];
            s += v * v;
        }
#pragma unroll
        for (int o = 16; o > 0; o >>= 1) s += __shfl_xor(s, o, 32);
        if (lane == 0) invB[n] = 1.0f / fmaxf(sqrtf(s), EPS_);
    }
    __syncthreads();

    // WMMA fp32 16x16x4:  A = tokens (16x768, rows 8..15 zero via invA==0)
    //                     B tiles  = cls_n^T (768x16) for n0=0 and n0=16
    const int  mrow = lane & 15;
    const int  kadd = (lane < 16) ? 0 : 2;       // ISA A/B 32-bit fragment layout
    const int  pA   = pOff[mrow];                // 0 for padded rows (valid addr)
    const float ia  = invA[mrow];                // 0 for padded rows
    const int  n1   = 16 + mrow;
    const float ib0 = invB[mrow];
    const float ib1 = invB[n1 & 31];             // 0 for n1 >= 21
    const float* aBase  = ptB + pA;
    const float* b0Base = clsB + mrow * C_;
    const float* b1Base = clsB + ((n1 < NCLS_) ? n1 : 0) * C_;

    v8f acc0 = {0.f, 0.f, 0.f, 0.f, 0.f, 0.f, 0.f, 0.f};
    v8f acc1 = {0.f, 0.f, 0.f, 0.f, 0.f, 0.f, 0.f, 0.f};

    for (int k0 = 0; k0 < C_; k0 += 4) {
        int c0 = k0 + kadd;
        v2f a, bb0, bb1;
        a.x   = aBase[(size_t)(c0    ) * HW_] * ia;   // branchless: ia==0 pads
        a.y   = aBase[(size_t)(c0 + 1) * HW_] * ia;
        bb0.x = b0Base[c0] * ib0;
        bb0.y = b0Base[c0 + 1] * ib0;
        bb1.x = b1Base[c0] * ib1;                     // ib1==0 pads cols >= 21
        bb1.y = b1Base[c0 + 1] * ib1;
        acc0 = __builtin_amdgcn_wmma_f32_16x16x4_f32(false, a, false, bb0,
                                                     (short)0, acc0, false, false);
        acc1 = __builtin_amdgcn_wmma_f32_16x16x4_f32(false, a, false, bb1,
                                                     (short)0, acc1, false, false);
    }

    // D layout: VGPR r -> row r (lanes 0-15, col=lane) and row r+8 (lanes 16-31,
    // col=lane-16). Padded rows/cols are exactly zero, so full-wave sums are exact.
    float t0[8], t1[8];
#pragma unroll
    for (int r = 0; r < 8; ++r) { t0[r] = acc0[r]; t1[r] = acc1[r]; }

    const float lab0 = labSh[mrow];
    const float lab1 = (n1 < NCLS_) ? labSh[n1] : 0.0f;
    const float bkg0 = (lab0 == 0.0f) ? 1.0f : 0.0f;
    const float bkg1 = (n1 < NCLS_) ? ((lab1 == 0.0f) ? 1.0f : 0.0f) : 0.0f;

    float lv = labSh[lane];                                   // 0 for lane >= 21
    float bv = (lane < NCLS_) ? ((lv == 0.0f) ? 1.0f : 0.0f) : 0.0f;
    float sl = lv, sb = bv;
#pragma unroll
    for (int o = 16; o > 0; o >>= 1) { sl += __shfl_xor(sl, o, 32); sb += __shfl_xor(sb, o, 32); }
    const float posDen = 1.0f / (sl + 1.0f);
    const float negDen = 1.0f / (sb + 1.0f);

    float lossLocal = 0.0f;
#pragma unroll
    for (int r = 0; r < 8; ++r) {
        float a0 = fabsf(t0[r]);
        float a1 = fabsf(t1[r]);
        float pp = lab0 * a0 + lab1 * a1;
        float nn = bkg0 * a0 + bkg1 * a1;
#pragma unroll
        for (int o = 16; o > 0; o >>= 1) { pp += __shfl_xor(pp, o, 32); nn += __shfl_xor(nn, o, 32); }
        lossLocal += 0.5f * (1.0f - pp * posDen) + 0.5f * (nn * negDen);
    }
    if (lane == 0) atomicAdd(out, lossLocal * (1.0f / (float)(B_ * CROP_)));
}

// ---------------------------------------------------------------------------
extern "C" void kernel_launch(void* const* d_in, const int* in_sizes, int n_in,
                              void* d_out, int out_size, void* d_ws, size_t ws_size,
                              hipStream_t stream) {
    (void)in_sizes; (void)n_in; (void)out_size; (void)ws_size;
    const float* cls   = (const float*)d_in[0];  // (B, 21, 768)
    const float* pt    = (const float*)d_in[1];  // (B, 768, 32, 32)
    const int*   masks = (const int*)d_in[2];    // (B, 32, 32)
    const float* lab   = (const float*)d_in[3];  // (B, 21)
    float* out = (float*)d_out;
    int*   sel = (int*)d_ws;                     // B*8 ints of scratch

    sre_select_kernel<<<B_, 256, 0, stream>>>(masks, sel, out);
    sre_gemm_kernel<<<B_, 32, 0, stream>>>(cls, pt, lab, sel, out);
}